// SAGE_VS_73555609911563
// MI455X (gfx1250) — compile-verified
//
#include <hip/hip_runtime.h>

#define NN 100000
#define EE 1250000
#define DD 64

typedef __attribute__((ext_vector_type(2))) float v2f;
typedef __attribute__((ext_vector_type(8))) float v8f;

// ---------------------------------------------------------------- zero fill
__global__ void zero_f32_vec4(float* __restrict__ p, long long n4) {
    long long t = (long long)blockIdx.x * blockDim.x + threadIdx.x;
    if (t < n4) {
        float4 z; z.x = 0.f; z.y = 0.f; z.z = 0.f; z.w = 0.f;
        ((float4*)p)[t] = z;
    }
}

// ---------------------------------------------------------------- in-degree
__global__ void count_deg(const int* __restrict__ dst, float* __restrict__ cnt) {
    int e = blockIdx.x * blockDim.x + threadIdx.x;
    if (e < EE) atomicAdd(&cnt[dst[e]], 1.0f);
}

// ------------------------------------------------- edge gather + scatter-add
// 16 threads per edge; each handles a float4 chunk of the 64-channel row.
__global__ void scatter_sum(const float* __restrict__ h,
                            const int* __restrict__ src,
                            const int* __restrict__ dst,
                            float* __restrict__ agg) {
    long long t = (long long)blockIdx.x * blockDim.x + threadIdx.x;
    int e  = (int)(t >> 4);
    int c4 = ((int)t & 15) * 4;
    if (e < EE) {
        int s = src[e];
        int d = dst[e];
        float4 v = *(const float4*)(h + (size_t)s * DD + c4);
        float* ap = agg + (size_t)d * DD + c4;
        atomicAdd(ap + 0, v.x);
        atomicAdd(ap + 1, v.y);
        atomicAdd(ap + 2, v.z);
        atomicAdd(ap + 3, v.w);
    }
}

// --------------------------------------------------------- fused SAGE layer
// out[n,:] = (agg[n,:]/max(cnt,1)) @ wl^T + b + h[n,:] @ wr^T  (+h, relu)
// Block: 128 threads = 4 waves; 64 nodes per block, 16 nodes per wave.
// All operands staged in LDS (CDNA5: 320 KB/WGP); the K-loop is pure
// ds_load + v_wmma_f32_16x16x4_f32, no global traffic, no exec divergence.
__launch_bounds__(128)
__global__ void sage_layer_wmma(const float* __restrict__ agg,
                                const float* __restrict__ cnt,
                                const float* __restrict__ h,
                                const float* __restrict__ wl,
                                const float* __restrict__ wr,
                                const float* __restrict__ bias,
                                float* __restrict__ out,
                                int relu_skip) {
    // stride-65 padding -> conflict-free stores, <=2-way-conflict frag loads
    __shared__ float wls[DD * 65];   // wls[o*65 + k] = wl[o][k]   (B = wl^T)
    __shared__ float wrs[DD * 65];
    __shared__ float ms [64 * 65];   // mean tile, ms[nl*65 + k]
    __shared__ float hs [64 * 65];   // h tile
    __shared__ float bs [DD];
    __shared__ float cs [64];

    const int tid   = threadIdx.x;
    const int node0 = blockIdx.x * 64;

    // stage weights + bias (coalesced reads; LDS stores hit distinct banks)
    for (int i = tid; i < DD * DD; i += 128) {
        int o = i >> 6, k = i & 63;
        wls[o * 65 + k] = wl[i];
        wrs[o * 65 + k] = wr[i];
    }
    if (tid < DD) bs[tid] = bias[tid];
    if (tid < 64) {
        int g = node0 + tid;
        cs[tid] = (g < NN) ? cnt[g] : 1.0f;
    }
    __syncthreads();

    // stage mean tile + h tile (coalesced along k)
    for (int i = tid; i < 64 * DD; i += 128) {
        int nl = i >> 6, k = i & 63;
        int g  = node0 + nl;
        float a = 0.0f, hv = 0.0f;
        if (g < NN) {
            a  = agg[(size_t)g * DD + k];
            hv = h[(size_t)g * DD + k];
        }
        ms[nl * 65 + k] = a * (1.0f / fmaxf(cs[nl], 1.0f));
        hs[nl * 65 + k] = hv;
    }
    __syncthreads();

    const int wave = tid >> 5;
    const int lane = tid & 31;
    const int half = lane >> 4;      // lane half selects K pair (ISA 16x4 layout)
    const int lo   = lane & 15;
    const int row  = wave * 16 + lo; // node-in-block for A fragments

    v8f acc[4] = {{}, {}, {}, {}};

#pragma unroll
    for (int kk = 0; kk < 16; ++kk) {
        const int k0 = kk * 4 + half * 2;   // A/B VGPR0:{K | K+2} per ISA layout
        v2f am, ah;
        am.x = ms[row * 65 + k0];
        am.y = ms[row * 65 + k0 + 1];
        ah.x = hs[row * 65 + k0];
        ah.y = hs[row * 65 + k0 + 1];

        v2f bl[4], br[4];
#pragma unroll
        for (int ot = 0; ot < 4; ++ot) {
            const int o = ot * 16 + lo;     // B row striped across lanes
            bl[ot].x = wls[o * 65 + k0];
            bl[ot].y = wls[o * 65 + k0 + 1];
            br[ot].x = wrs[o * 65 + k0];
            br[ot].y = wrs[o * 65 + k0 + 1];
        }
        // sweep all 4 tiles with A=mean, then all 4 with A=h: any two WMMAs
        // sharing an accumulator are 4 instructions apart (hides XDL latency)
#pragma unroll
        for (int ot = 0; ot < 4; ++ot)
            acc[ot] = __builtin_amdgcn_wmma_f32_16x16x4_f32(
                false, am, false, bl[ot], (short)0, acc[ot], false, false);
#pragma unroll
        for (int ot = 0; ot < 4; ++ot)
            acc[ot] = __builtin_amdgcn_wmma_f32_16x16x4_f32(
                false, ah, false, br[ot], (short)0, acc[ot], false, false);
    }

    // epilogue: C/D layout -> VGPR r holds M=r (lanes 0-15), M=r+8 (lanes 16-31)
#pragma unroll
    for (int ot = 0; ot < 4; ++ot) {
        const int o = ot * 16 + lo;
#pragma unroll
        for (int r = 0; r < 8; ++r) {
            int nl = wave * 16 + r + half * 8;
            int g  = node0 + nl;
            if (g < NN) {
                float v = acc[ot][r] + bs[o];
                if (relu_skip) {
                    v += hs[nl * 65 + o];         // residual (from LDS)
                    v  = fmaxf(v, 0.0f);          // relu
                }
                out[(size_t)g * DD + o] = v;
            }
        }
    }
}

// ---------------------------------------------------------------- launcher
extern "C" void kernel_launch(void* const* d_in, const int* in_sizes, int n_in,
                              void* d_out, int out_size, void* d_ws, size_t ws_size,
                              hipStream_t stream) {
    const float* x  = (const float*)d_in[0];
    const int*   ei = (const int*)d_in[1];    // [2, E] row-major
    const float* wl = (const float*)d_in[2];  // [L, D, D]
    const float* wr = (const float*)d_in[3];  // [L, D, D]
    const float* b  = (const float*)d_in[4];  // [L, D]
    const int* src = ei;
    const int* dst = ei + EE;

    // workspace carve-out: h0 | h1 | agg | cnt  (~77 MB)
    float* h0  = (float*)d_ws;
    float* h1  = h0 + (size_t)NN * DD;
    float* agg = h1 + (size_t)NN * DD;
    float* cnt = agg + (size_t)NN * DD;

    // seed h0 = x (inputs must not be mutated)
    hipMemcpyAsync(h0, x, (size_t)NN * DD * sizeof(float),
                   hipMemcpyDeviceToDevice, stream);

    // in-degrees (layer-invariant): zero then atomic count
    {
        long long n4 = NN / 4;
        zero_f32_vec4<<<(unsigned)((n4 + 255) / 256), 256, 0, stream>>>(cnt, n4);
        count_deg<<<(EE + 255) / 256, 256, 0, stream>>>(dst, cnt);
    }

    float* cur = h0;
    float* nxt = h1;
    const unsigned gemm_blocks = (NN + 63) / 64;

    for (int layer = 0; layer < 4; ++layer) {
        long long n4 = (long long)NN * DD / 4;
        zero_f32_vec4<<<(unsigned)((n4 + 255) / 256), 256, 0, stream>>>(agg, n4);

        long long st = (long long)EE * 16;
        scatter_sum<<<(unsigned)((st + 255) / 256), 256, 0, stream>>>(cur, src, dst, agg);

        const float* wli = wl + (size_t)layer * DD * DD;
        const float* wri = wr + (size_t)layer * DD * DD;
        const float* bi  = b + (size_t)layer * DD;
        float* o = (layer == 3) ? (float*)d_out : nxt;

        sage_layer_wmma<<<gemm_blocks, 128, 0, stream>>>(
            agg, cnt, cur, wli, wri, bi, o, (layer < 3) ? 1 : 0);

        float* t = cur; cur = nxt; nxt = t;
    }
}